// AttnDecoder_78640851190326
// MI455X (gfx1250) — compile-verified
//
#include <hip/hip_runtime.h>
#include <hip/hip_bf16.h>
#include <math.h>

#define H 1024
#define V 50257
#define S 8192
#define TILE_STRIDE 68   // 16-row LDS tile stride (floats): 16B aligned rows, bank-conflict free

typedef __attribute__((ext_vector_type(2))) float v2f;
typedef __attribute__((ext_vector_type(8))) float v8f;
typedef __attribute__((ext_vector_type(4))) int   v4i;
typedef __attribute__((address_space(1))) v4i* gv4i_p;  // global int4*
typedef __attribute__((address_space(3))) v4i* lv4i_p;  // LDS int4*

#if defined(__has_builtin)
#if __has_builtin(__builtin_amdgcn_global_load_async_to_lds_b128)
#define HAVE_ASYNC_LDS 1
#endif
#endif

#ifndef HAVE_ASYNC_LDS
#define HAVE_ASYNC_LDS 0
#endif

#if defined(__has_builtin)
#if __has_builtin(__builtin_amdgcn_s_wait_asynccnt)
#define WAIT_ASYNC(n) __builtin_amdgcn_s_wait_asynccnt(n)
#else
#define WAIT_ASYNC(n) asm volatile("s_wait_asynccnt " #n ::: "memory")
#endif
#else
#define WAIT_ASYNC(n) asm volatile("s_wait_asynccnt " #n ::: "memory")
#endif

// ---------------------------------------------------------------------------
// Kernel 1: gi = W_ih @ x + b_ih   (x = [emb[word], last_context], 2048)
//           gh = W_hh @ h + b_hh   (h = last_hidden, 1024)
// One block (256 thr) per output row; coalesced contiguous per-thread chunks.
// ---------------------------------------------------------------------------
__global__ __launch_bounds__(256) void gates_kernel(
    const int* __restrict__ word, const float* __restrict__ last_context,
    const float* __restrict__ last_hidden, const float* __restrict__ embedding,
    const float* __restrict__ W_ih, const float* __restrict__ W_hh,
    const float* __restrict__ b_ih, const float* __restrict__ b_hh,
    float* __restrict__ gi, float* __restrict__ gh) {
  const int b   = blockIdx.x;
  const int tid = threadIdx.x;
  float partial = 0.f;
  if (b < 3 * H) {
    const float* wrow = W_ih + (size_t)b * (2 * H);
    const float* embr = embedding + (size_t)word[0] * H;
    const int k0 = tid * 8;
#pragma unroll
    for (int u = 0; u < 8; ++u) {
      const int k = k0 + u;
      const float xv = (k < H) ? embr[k] : last_context[k - H];
      partial += xv * wrow[k];
    }
  } else {
    const int r = b - 3 * H;
    const float* wrow = W_hh + (size_t)r * H;
    const int k0 = tid * 4;
#pragma unroll
    for (int u = 0; u < 4; ++u) partial += last_hidden[k0 + u] * wrow[k0 + u];
  }
  for (int off = 16; off > 0; off >>= 1) partial += __shfl_down(partial, off, 32);
  __shared__ float red[8];
  const int lane = tid & 31, w = tid >> 5;
  if (lane == 0) red[w] = partial;
  __syncthreads();
  if (tid == 0) {
    float s = 0.f;
#pragma unroll
    for (int i = 0; i < 8; ++i) s += red[i];
    if (b < 3 * H) gi[b] = s + b_ih[b];
    else           gh[b - 3 * H] = s + b_hh[b - 3 * H];
  }
}

// ---------------------------------------------------------------------------
// Kernel 2: GRU gate math -> h_new (1024)
// ---------------------------------------------------------------------------
__global__ __launch_bounds__(1024) void gru_kernel(
    const float* __restrict__ gi, const float* __restrict__ gh,
    const float* __restrict__ last_hidden, float* __restrict__ hnew) {
  const int j = threadIdx.x;
  const float r = 1.f / (1.f + expf(-(gi[j] + gh[j])));
  const float z = 1.f / (1.f + expf(-(gi[H + j] + gh[H + j])));
  const float n = tanhf(gi[2 * H + j] + r * gh[2 * H + j]);
  hnew[j] = (1.f - z) * n + z * last_hidden[j];
}

// ---------------------------------------------------------------------------
// Kernel 3: scores[s] = dot(enc[s], h_new). One block per row, float4 loads.
// ---------------------------------------------------------------------------
__global__ __launch_bounds__(256) void scores_kernel(
    const float* __restrict__ enc, const float* __restrict__ hnew,
    float* __restrict__ scores) {
  const int s = blockIdx.x, tid = threadIdx.x;
  const float4 a = ((const float4*)(enc + (size_t)s * H))[tid];
  const float4 h = ((const float4*)hnew)[tid];
  float partial = a.x * h.x + a.y * h.y + a.z * h.z + a.w * h.w;
  for (int off = 16; off > 0; off >>= 1) partial += __shfl_down(partial, off, 32);
  __shared__ float red[8];
  const int lane = tid & 31, w = tid >> 5;
  if (lane == 0) red[w] = partial;
  __syncthreads();
  if (tid == 0) {
    float sum = 0.f;
#pragma unroll
    for (int i = 0; i < 8; ++i) sum += red[i];
    scores[s] = sum;
  }
}

// ---------------------------------------------------------------------------
// Kernel 4: softmax over 8192 scores -> attn; also zero the context output
// (single block, 1024 threads, 8 values per thread kept in registers)
// ---------------------------------------------------------------------------
__global__ __launch_bounds__(1024) void softmax_kernel(
    const float* __restrict__ scores, float* __restrict__ attn,
    float* __restrict__ ctx_zero) {
  const int tid = threadIdx.x;
  const int lane = tid & 31, w = tid >> 5;
  __shared__ float redm[32];
  __shared__ float reds[32];
  float v[8];
  float m = -INFINITY;
#pragma unroll
  for (int i = 0; i < 8; ++i) { v[i] = scores[tid + i * 1024]; m = fmaxf(m, v[i]); }
  for (int off = 16; off > 0; off >>= 1) m = fmaxf(m, __shfl_down(m, off, 32));
  if (lane == 0) redm[w] = m;
  __syncthreads();
  if (w == 0) {
    float t = redm[lane];
    for (int off = 16; off > 0; off >>= 1) t = fmaxf(t, __shfl_down(t, off, 32));
    if (lane == 0) redm[0] = t;
  }
  __syncthreads();
  m = redm[0];
  float s = 0.f;
#pragma unroll
  for (int i = 0; i < 8; ++i) { v[i] = expf(v[i] - m); s += v[i]; }
  for (int off = 16; off > 0; off >>= 1) s += __shfl_down(s, off, 32);
  if (lane == 0) reds[w] = s;
  __syncthreads();
  if (w == 0) {
    float t = reds[lane];
    for (int off = 16; off > 0; off >>= 1) t += __shfl_down(t, off, 32);
    if (lane == 0) reds[0] = t;
  }
  __syncthreads();
  const float inv = 1.f / reds[0];
#pragma unroll
  for (int i = 0; i < 8; ++i) attn[tid + i * 1024] = v[i] * inv;
  ctx_zero[tid] = 0.f;  // 1024 threads zero the 1024-float context accumulator
}

// ---------------------------------------------------------------------------
// Kernel 5: context[c] = sum_s attn[s] * enc[s, c]. Grid (64 row-grps, 4 col-grps),
// coalesced column reads, one f32 atomicAdd per thread per row-group.
// ---------------------------------------------------------------------------
__global__ __launch_bounds__(256) void context_kernel(
    const float* __restrict__ attn, const float* __restrict__ enc,
    float* __restrict__ ctx) {
  const int col = blockIdx.y * 256 + threadIdx.x;
  const int s0  = blockIdx.x * 128;
  float acc = 0.f;
  for (int s = s0; s < s0 + 128; ++s) acc += attn[s] * enc[(size_t)s * H + col];
  atomicAdd(ctx + col, acc);
}

// ---------------------------------------------------------------------------
// Kernel 6: logits = W_out @ [h_new, context] + b_out via V_WMMA_F32_16X16X4_F32.
// 8 waves/block; each wave owns 16 rows of W_out.
//
// Async path (gfx1250): double-buffered GLOBAL_LOAD_ASYNC_TO_LDS_B128 staging of
// wave-private 16x64 tiles; s_wait_asynccnt 8 pipelines copy vs. WMMA (loads
// complete in order, 8 per tile -> waiting to <=8 outstanding resolves the
// previous tile exactly). Rows >= V are clamped to row V-1 so the per-tile
// async-op count is always 8; duplicate data lands only in M-rows never stored.
//
// WMMA operand layout (ISA f32 16x16x4): lanes 0-15 hold K=k+0/k+1 in VGPR0/1,
// lanes 16-31 hold K=k+2/k+3. B = x replicated across all 16 N columns, so
// every D column equals the matvec; lanes 0/16 extract N=0 for M=0..7 / 8..15.
// ---------------------------------------------------------------------------
__global__ __launch_bounds__(256) void logits_wmma_kernel(
    const float* __restrict__ hnew, const float* __restrict__ ctx,
    const float* __restrict__ W_out, const float* __restrict__ b_out,
    float* __restrict__ logits) {
  __shared__ float xs[2 * H];
  const int tid  = threadIdx.x;
  const int lane = tid & 31;
  const int w    = tid >> 5;

  for (int k = tid; k < 2 * H; k += 256)
    xs[k] = (k < H) ? hnew[k] : ctx[k - H];
  __syncthreads();

  const int row_base = blockIdx.x * 128 + w * 16;
  v8f c = {0.f, 0.f, 0.f, 0.f, 0.f, 0.f, 0.f, 0.f};
  const int koff = (lane < 16) ? 0 : 2;   // K split across lane halves (ISA layout)
  const int arow = lane & 15;

#if HAVE_ASYNC_LDS
  __shared__ float tiles[8][2][16 * TILE_STRIDE];

  // stage one wave-private 16x64 tile with 8 async b128 loads (VGPR-free)
  auto stage = [&](int kc, int buf) {
#pragma unroll
    for (int i = 0; i < 8; ++i) {
      const int idx = i * 32 + lane;       // 0..255
      const int r   = idx >> 4;            // 0..15
      const int cc  = (idx & 15) * 4;      // 0..60
      int grow = row_base + r;
      if (grow >= V) grow = V - 1;         // clamp: keep async-op count uniform
      __builtin_amdgcn_global_load_async_to_lds_b128(
          (gv4i_p)(const_cast<float*>(W_out) + (size_t)grow * (2 * H) + kc + cc),
          (lv4i_p)&tiles[w][buf][r * TILE_STRIDE + cc],
          0, 0);
    }
  };

  stage(0, 0);
  for (int kc = 0; kc < 2 * H; kc += 64) {
    const int cur = (kc >> 6) & 1;
    if (kc + 64 < 2 * H) {
      stage(kc + 64, cur ^ 1);
      WAIT_ASYNC(8);                       // previous tile's 8 loads resolved
    } else {
      WAIT_ASYNC(0);
    }
    asm volatile("" ::: "memory");         // keep ds_loads below the wait
#pragma unroll
    for (int k4 = 0; k4 < 64; k4 += 4) {
      const float* xp = &xs[kc + k4 + koff];
      v2f bv; bv.x = xp[0]; bv.y = xp[1];
      const float* ap = &tiles[w][cur][arow * TILE_STRIDE + k4 + koff];
      v2f av; av.x = ap[0]; av.y = ap[1];
      c = __builtin_amdgcn_wmma_f32_16x16x4_f32(
          false, av, false, bv, (short)0, c, false, false);
    }
  }
#else
  __shared__ float tiles[8][16 * TILE_STRIDE];

  for (int kc = 0; kc < 2 * H; kc += 64) {
#pragma unroll
    for (int i = 0; i < 8; ++i) {
      const int idx = i * 32 + lane;
      const int r   = idx >> 4;
      const int cc  = (idx & 15) * 4;
      const int grow = row_base + r;
      float4 val = make_float4(0.f, 0.f, 0.f, 0.f);
      if (grow < V)
        val = *(const float4*)(W_out + (size_t)grow * (2 * H) + kc + cc);
      *(float4*)&tiles[w][r * TILE_STRIDE + cc] = val;
    }
    __syncthreads();
#pragma unroll
    for (int k4 = 0; k4 < 64; k4 += 4) {
      const float* xp = &xs[kc + k4 + koff];
      v2f bv; bv.x = xp[0]; bv.y = xp[1];
      const float* ap = &tiles[w][arow * TILE_STRIDE + k4 + koff];
      v2f av; av.x = ap[0]; av.y = ap[1];
      c = __builtin_amdgcn_wmma_f32_16x16x4_f32(
          false, av, false, bv, (short)0, c, false, false);
    }
    __syncthreads();
  }
#endif

  if (lane == 0 || lane == 16) {
    const int mbase = (lane == 0) ? 0 : 8;
#pragma unroll
    for (int j = 0; j < 8; ++j) {
      const int row = row_base + mbase + j;
      if (row < V) logits[row] = c[j] + b_out[row];
    }
  }
}

// ---------------------------------------------------------------------------
// Kernel 7: in-place log_softmax over the 50257 logits (single block).
// ---------------------------------------------------------------------------
__global__ __launch_bounds__(1024) void logsoftmax_kernel(float* __restrict__ out) {
  const int tid = threadIdx.x;
  const int lane = tid & 31, w = tid >> 5;
  __shared__ float red[32];
  float m = -INFINITY;
  for (int i = tid; i < V; i += 1024) m = fmaxf(m, out[i]);
  for (int off = 16; off > 0; off >>= 1) m = fmaxf(m, __shfl_down(m, off, 32));
  if (lane == 0) red[w] = m;
  __syncthreads();
  if (w == 0) {
    float t = red[lane];
    for (int off = 16; off > 0; off >>= 1) t = fmaxf(t, __shfl_down(t, off, 32));
    if (lane == 0) red[0] = t;
  }
  __syncthreads();
  m = red[0];
  __syncthreads();
  float s = 0.f;
  for (int i = tid; i < V; i += 1024) s += expf(out[i] - m);
  for (int off = 16; off > 0; off >>= 1) s += __shfl_down(s, off, 32);
  if (lane == 0) red[w] = s;
  __syncthreads();
  if (w == 0) {
    float t = red[lane];
    for (int off = 16; off > 0; off >>= 1) t += __shfl_down(t, off, 32);
    if (lane == 0) red[0] = t;
  }
  __syncthreads();
  const float L = m + logf(red[0]);
  for (int i = tid; i < V; i += 1024) out[i] -= L;
}

// ---------------------------------------------------------------------------
extern "C" void kernel_launch(void* const* d_in, const int* in_sizes, int n_in,
                              void* d_out, int out_size, void* d_ws, size_t ws_size,
                              hipStream_t stream) {
  const int*   word      = (const int*)d_in[0];
  const float* last_ctx  = (const float*)d_in[1];
  const float* last_hid  = (const float*)d_in[2];
  const float* enc       = (const float*)d_in[3];
  const float* embedding = (const float*)d_in[4];
  const float* W_ih      = (const float*)d_in[5];
  const float* W_hh      = (const float*)d_in[6];
  const float* b_ih      = (const float*)d_in[7];
  const float* b_hh      = (const float*)d_in[8];
  const float* W_out     = (const float*)d_in[9];
  const float* b_out     = (const float*)d_in[10];

  // d_out layout (reference tuple, flat): logits | context | h_new | attn
  float* out_logits = (float*)d_out;
  float* out_ctx    = out_logits + V;
  float* out_hnew   = out_ctx + H;
  float* out_attn   = out_hnew + H;

  // workspace: gi(3072) | gh(3072) | scores(8192)
  float* gi     = (float*)d_ws;
  float* gh     = gi + 3 * H;
  float* scores = gh + 3 * H;

  gates_kernel<<<6 * H, 256, 0, stream>>>(word, last_ctx, last_hid, embedding,
                                          W_ih, W_hh, b_ih, b_hh, gi, gh);
  gru_kernel<<<1, 1024, 0, stream>>>(gi, gh, last_hid, out_hnew);
  scores_kernel<<<S, 256, 0, stream>>>(enc, out_hnew, scores);
  softmax_kernel<<<1, 1024, 0, stream>>>(scores, out_attn, out_ctx);
  context_kernel<<<dim3(64, 4), 256, 0, stream>>>(out_attn, enc, out_ctx);
  logits_wmma_kernel<<<(V + 127) / 128, 256, 0, stream>>>(out_hnew, out_ctx,
                                                          W_out, b_out, out_logits);
  logsoftmax_kernel<<<1, 1024, 0, stream>>>(out_logits);
}